// NewNCutLoss_28578712388234
// MI455X (gfx1250) — compile-verified
//
#include <hip/hip_runtime.h>

typedef __attribute__((ext_vector_type(2))) float v2f;
typedef __attribute__((ext_vector_type(8))) float v8f;

// One pass over edges. Each wave processes 4 edges per WMMA:
//   A[g,k] = w_k * Y[src_k, g]   (16x4 f32, M=g, K=edge-in-tile)
//   B[k,n] = 1 - Y[dst_k, n]     (4x16 f32, K=edge-in-tile, N=n)
//   C += A x B   -> Num[g] accumulates on diag(C)
//   Gamma[g] accumulates as the lane-local sum of A elements.
// f32 A layout (ISA 7.12.2): lanes 0-15 hold K={0,1} (VGPR0,VGPR1), lanes
// 16-31 hold K={2,3}; M = lane&15. B mirrors the lane-half K split with
// N = lane&15. Every operand fill is a coalesced 16-lane 64B row gather.
template <bool GUARD>
__device__ __forceinline__ void tile4(
    const float* __restrict__ Y, const float* __restrict__ ew,
    const int* __restrict__ eiSrc, const int* __restrict__ eiDst,
    int E, int base, int h, int g, v8f& c, float& gammaAcc)
{
    int eA = base + 2 * h;      // edge feeding VGPR0 of A/B in this half
    int eB = eA + 1;            // edge feeding VGPR1 of A/B in this half
    int srcA, srcB, dstA, dstB;
    float wA, wB;
    if (GUARD) {
        bool vA = eA < E, vB = eB < E;
        int cAi = vA ? eA : 0;
        int cBi = vB ? eB : 0;
        srcA = eiSrc[cAi]; srcB = eiSrc[cBi];
        dstA = eiDst[cAi]; dstB = eiDst[cBi];
        wA = ew[cAi]; if (!vA) wA = 0.0f;   // w=0 kills invalid-edge terms
        wB = ew[cBi]; if (!vB) wB = 0.0f;
    } else {
        srcA = eiSrc[eA]; srcB = eiSrc[eB];
        dstA = eiDst[eA]; dstB = eiDst[eB];
        wA = ew[eA]; wB = ew[eB];
    }
    float a0 = wA * Y[srcA * 16 + g];       // A[M=g, K=2h]
    float a1 = wB * Y[srcB * 16 + g];       // A[M=g, K=2h+1]
    float b0 = 1.0f - Y[dstA * 16 + g];     // B[K=2h,   N=g]
    float b1 = 1.0f - Y[dstB * 16 + g];     // B[K=2h+1, N=g]
    gammaAcc += a0 + a1;                    // Gamma_g partial (A row sums)
    v2f a; a[0] = a0; a[1] = a1;
    v2f b; b[0] = b0; b[1] = b1;
    c = __builtin_amdgcn_wmma_f32_16x16x4_f32(false, a, false, b,
                                              (short)0, c, false, false);
}

__global__ __launch_bounds__(256) void ncut_edge_partials(
    const float* __restrict__ Y,
    const float* __restrict__ ew,
    const int*   __restrict__ eiSrc,  // ei
    const int*   __restrict__ eiDst,  // ei + E
    float* __restrict__ part,         // [gridDim.x * 32] block partials
    int E)
{
    __shared__ float sPart[8][32];

    const int tid  = threadIdx.x;
    const int lane = tid & 31;
    const int wv   = tid >> 5;
    const int h    = lane >> 4;     // lane half -> K pair {2h, 2h+1}
    const int g    = lane & 15;     // M index for A, N index for B

    const int waveId = blockIdx.x * 8 + wv;
    const int stride = gridDim.x * 8 * 8;   // 8 edges per wave per step

    v8f   c0 = {}, c1 = {};         // two independent 16x16 f32 accumulators
    float gammaAcc = 0.0f;

    int base = waveId * 8;
    // Main loop: full 8-edge steps, branch-free tiles.
    for (; base + 8 <= E; base += stride) {
        tile4<false>(Y, ew, eiSrc, eiDst, E, base,     h, g, c0, gammaAcc);
        tile4<false>(Y, ew, eiSrc, eiDst, E, base + 4, h, g, c1, gammaAcc);
    }
    // Tail: at most one partial step per wave; conditions are wave-uniform,
    // so EXEC is all-1s at each WMMA.
    if (base < E) {
        tile4<true>(Y, ew, eiSrc, eiDst, E, base, h, g, c0, gammaAcc);
        if (base + 4 < E)
            tile4<true>(Y, ew, eiSrc, eiDst, E, base + 4, h, g, c1, gammaAcc);
    }
    c0 += c1;

    // Extract diag(C). C/D layout: VGPR r, lane l -> (M = r + 8*(l>=16), N = l&15).
    // diag (q,q): q<8 -> lane q, VGPR q ; q>=8 -> lane q+16, VGPR q-8.
    float diag = 0.0f;
    int r = -1;
    if (lane < 8)        r = lane;
    else if (lane >= 24) r = lane - 24;
#pragma unroll
    for (int i = 0; i < 8; ++i)
        if (r == i) diag = c0[i];

    // Gather so lanes 0..15 hold Num[lane] (Num[8..15] live in lanes 24..31).
    int sl = (lane < 8) ? lane : lane + 16;
    float numG = __shfl(diag, sl, 32);

    // Gamma[g] = lane g half + lane g+16 half.
    float gamG = gammaAcc + __shfl_down(gammaAcc, 16, 32);

    if (lane < 16) {
        sPart[wv][lane]      = gamG;
        sPart[wv][16 + lane] = numG;
    }
    __syncthreads();

    // Ordered (deterministic) cross-wave sum, one partial record per block.
    if (tid < 32) {
        float s = 0.0f;
#pragma unroll
        for (int k = 0; k < 8; ++k) s += sPart[k][tid];
        part[blockIdx.x * 32 + tid] = s;
    }
}

// Deterministic serial reduce of block partials, then loss = sum_g Num_g/Gamma_g.
__global__ void ncut_finalize(const float* __restrict__ part, int nb,
                              float* __restrict__ out)
{
    int t = threadIdx.x;            // 32 threads, one wave
    float acc = 0.0f;
    for (int b = 0; b < nb; ++b) acc += part[b * 32 + t];
    // t in [0,16): Gamma[t] ; t in [16,32): Num[t-16]
    float gam = __shfl(acc, t & 15, 32);
    float v = (t >= 16) ? (acc / gam) : 0.0f;
    for (int off = 16; off > 0; off >>= 1) v += __shfl_down(v, off, 32);
    if (t == 0) out[0] = v;
}

extern "C" void kernel_launch(void* const* d_in, const int* in_sizes, int n_in,
                              void* d_out, int out_size, void* d_ws, size_t ws_size,
                              hipStream_t stream) {
    (void)n_in; (void)out_size;
    const float* Y  = (const float*)d_in[0];
    const float* ew = (const float*)d_in[1];
    const int*   ei = (const int*)d_in[2];
    const int E = in_sizes[1];

    int blocks = 512;               // 4096 waves, ~98 dual-WMMA steps each
    size_t need = (size_t)blocks * 32 * sizeof(float);
    if (need > ws_size) {
        blocks = (int)(ws_size / (32 * sizeof(float)));
        if (blocks < 1) blocks = 1;
    }
    float* part = (float*)d_ws;

    ncut_edge_partials<<<blocks, 256, 0, stream>>>(Y, ew, ei, ei + E, part, E);
    ncut_finalize<<<1, 32, 0, stream>>>(part, blocks, (float*)d_out);
}